// WAVE_34900904247805
// MI455X (gfx1250) — compile-verified
//
#include <hip/hip_runtime.h>

typedef __attribute__((ext_vector_type(2))) float v2f;
typedef __attribute__((ext_vector_type(8))) float v8f;

#define FD 64
#define PD 64
#define LV 16
#define STRIDE 68          // padded row stride (floats) to avoid LDS bank conflicts
#define FS (FD * STRIDE)   // 4352 floats per 64x64 padded matrix (4352 % 64 == 0)
#define NT 256

__launch_bounds__(256, 1)
__global__ void WAVE_34900904247805_kernel(
    const float* __restrict__ x,
    const float* __restrict__ W_TA, const float* __restrict__ b_TA,
    const float* __restrict__ W_CA, const float* __restrict__ b_CA,
    const float* __restrict__ W_TB, const float* __restrict__ b_TB,
    const float* __restrict__ W_CB, const float* __restrict__ b_CB,
    const float* __restrict__ W_mix,
    const float* __restrict__ W_u, const float* __restrict__ b_u,
    const float* __restrict__ W_o, const float* __restrict__ b_o,
    const float* __restrict__ W1, const float* __restrict__ b1,
    const float* __restrict__ W2, const float* __restrict__ b2,
    const int* __restrict__ num_passes_p,
    float* __restrict__ ws,        // 16*64*64 floats: states (L, F, P)
    float* __restrict__ out)       // (OUT=16, N=1024)
{
  // [0]=C, [1]=T  (merged so selection is an integer base offset, not a branch)
  __shared__ float sCT[2 * FS];
  // 0=YTA, 1=YTB, 2=YCA, 3=YCB
  __shared__ float sY[4 * FS];
  // transposed weight tables for inner-loop mat-vecs (k-major, lane-consecutive r)
  __shared__ float sWRT[64 * 256];    // [k][r]: r<64 TA_R, <128 CA_R, <192 TB_R, else CB_R
  __shared__ float sWUOT[128 * 128];  // [k][r]: r<64 W_u rows, else W_o rows
  __shared__ float sWCLT[64 * 128];   // [k][r]: r<64 W_CA_L rows, else W_CB_L rows
  __shared__ float sVB[256];          // stacked biases bTA,bCA,bTB,bCB
  __shared__ float sBUO[128];         // b_u, b_o
  __shared__ float sWMIX[64];
  __shared__ float sVbuf[256];        // per-iteration rank-1 terms (+bias)
  __shared__ float sMix[64];
  __shared__ float sG[128];           // raw u/o pre-activations
  __shared__ float sNew[64];          // new column

  const int tid  = threadIdx.x;
  const int lane = tid & 31;
  const int wave = tid >> 5;

  // ---------------- init LDS weight tables ----------------
  for (int idx = tid; idx < 64 * 256; idx += NT) {
    int k = idx >> 8, r = idx & 255, g = r >> 6, rr = r & 63;
    const float* W = (g == 0) ? W_TA : (g == 1) ? W_CA : (g == 2) ? W_TB : W_CB;
    sWRT[idx] = W[rr * 128 + 64 + k];
  }
  for (int idx = tid; idx < 128 * 128; idx += NT) {
    int k = idx >> 7, r = idx & 127;
    sWUOT[idx] = (r < 64) ? W_u[r * 128 + k] : W_o[(r - 64) * 128 + k];
  }
  for (int idx = tid; idx < 64 * 128; idx += NT) {
    int k = idx >> 7, r = idx & 127;
    sWCLT[idx] = (r < 64) ? W_CA[r * 128 + k] : W_CB[(r - 64) * 128 + k];
  }
  {
    int g = tid >> 6, rr = tid & 63;
    sVB[tid] = (g == 0) ? b_TA[rr] : (g == 1) ? b_CA[rr] : (g == 2) ? b_TB[rr] : b_CB[rr];
    if (tid < 128) sBUO[tid] = (tid < 64) ? b_u[tid] : b_o[tid - 64];
    if (tid < 64)  sWMIX[tid] = W_mix[tid];
  }
  // states = transpose(x, (0,2,1)) : ws[l][f][p] = x[l][p][f]
  for (int idx = tid; idx < LV * FD * PD; idx += NT) {
    int l = idx >> 12, rem = idx & 4095, f = rem >> 6, p = rem & 63;
    ws[idx] = x[l * 4096 + p * 64 + f];
  }
  __syncthreads();

  const int np = num_passes_p[0];

  for (int pass = 0; pass < np; ++pass) {
    #pragma unroll 1
    for (int sweep = 0; sweep < 2; ++sweep) {
      #pragma unroll 1
      for (int step = 0; step < LV - 1; ++step) {
        const int lc = (sweep == 0) ? (1 + step) : (LV - 2 - step);
        const int lt = (sweep == 0) ? (lc - 1) : (lc + 1);

        // ---- load active levels into LDS ----
        for (int idx = tid; idx < FD * PD; idx += NT) {
          int f = idx >> 6, p = idx & 63;
          sCT[f * STRIDE + p]      = ws[lc * 4096 + idx];
          sCT[FS + f * STRIDE + p] = ws[lt * 4096 + idx];
        }
        __syncthreads();

        // ---- Phase A: 4 hoisted GEMMs (64x64x64, fp32 WMMA) ----
        // g: 0 -> YTA=W_TA_L@T, 1 -> YTB=W_TB_L@T, 2 -> YCA=W_CA_L@C, 3 -> YCB=W_CB_L@C
        {
          const int M  = lane & 15;
          const int hi = lane >> 4;           // 0/1 half-lane group
          const int g  = wave >> 1;           // wave-uniform
          const float* Wp = (g == 0) ? W_TA : (g == 1) ? W_TB : (g == 2) ? W_CA : W_CB;
          const int bbase = (g < 2) ? FS : 0; // B operand: T for g<2, else C (no branch)
          const int ybase = g * FS;
          #pragma unroll
          for (int jj = 0; jj < 8; ++jj) {
            const int t  = (wave & 1) * 8 + jj;
            const int mt = t >> 2, nt = t & 3;
            v8f d = {0.f, 0.f, 0.f, 0.f, 0.f, 0.f, 0.f, 0.f};
            #pragma unroll
            for (int kt = 0; kt < 16; ++kt) {
              const int k0 = kt * 4 + hi * 2;
              v2f a, b;
              a.x = Wp[(mt * 16 + M) * 128 + k0];
              a.y = Wp[(mt * 16 + M) * 128 + k0 + 1];
              b.x = sCT[bbase + k0 * STRIDE + nt * 16 + M];
              b.y = sCT[bbase + (k0 + 1) * STRIDE + nt * 16 + M];
              d = __builtin_amdgcn_wmma_f32_16x16x4_f32(
                      false, a, false, b, (short)0, d, false, false);
            }
            const int row0 = mt * 16 + hi * 8;
            const int col  = nt * 16 + M;
            #pragma unroll
            for (int r = 0; r < 8; ++r)
              sY[ybase + (row0 + r) * STRIDE + col] = d[r];
          }
        }
        __syncthreads();

        // ---- Phase B: sequential scan over cross nodes ----
        #pragma unroll 1
        for (int i = 0; i < PD; ++i) {
          // step 2: rank-1 terms v = W_*R @ s (+bias), s = old C[:,i]
          {
            float acc = sVB[tid];
            #pragma unroll 8
            for (int k = 0; k < 64; ++k)
              acc = fmaf(sWRT[k * 256 + tid], sCT[k * STRIDE + i], acc);
            sVbuf[tid] = acc;
          }
          __syncthreads();

          // step 3: per-row softmax/softsign/mix over 128 neighbors (4 lanes/row),
          // fully branchless via computed LDS base offsets.
          {
            const int r   = tid >> 2, q = tid & 3;
            const int hiQ = q >> 1;                // 0: T-derived half, 1: C-derived half
            const int pb  = (q & 1) * 32;
            const int zaB = hiQ * 2 * FS + r * STRIDE + pb;   // YTA or YCA
            const int zbB = zaB + FS;                         // YTB or YCB
            const int nbB = hiQ * FS + r * STRIDE + pb;       // Nbr = [C, T]
            const float vA = sVbuf[hiQ * 64 + r];
            const float vB = sVbuf[128 + hiQ * 64 + r];
            float za[32];
            float mx = -3.4e38f;
            #pragma unroll
            for (int e = 0; e < 32; ++e) {
              za[e] = sY[zaB + e] + vA;
              mx = fmaxf(mx, za[e]);
            }
            mx = fmaxf(mx, __shfl_xor(mx, 1));
            mx = fmaxf(mx, __shfl_xor(mx, 2));
            float es = 0.f, wsum = 0.f, ssum = 0.f;
            #pragma unroll
            for (int e = 0; e < 32; ++e) {
              const float ex = __expf(za[e] - mx);
              const float nb = sCT[nbB + e];
              es += ex; wsum += ex * nb;
              const float zb = sY[zbB + e] + vB;
              ssum += (zb / (1.f + fabsf(zb))) * nb;
            }
            es   += __shfl_xor(es, 1);   es   += __shfl_xor(es, 2);
            wsum += __shfl_xor(wsum, 1); wsum += __shfl_xor(wsum, 2);
            ssum += __shfl_xor(ssum, 1); ssum += __shfl_xor(ssum, 2);
            if (q == 0) sMix[r] = sWMIX[r] * (wsum / es) + ssum;
          }
          __syncthreads();

          // step 4: gate pre-activations, cat = [mix; s]
          if (tid < 128) {
            float acc = sBUO[tid];
            #pragma unroll 8
            for (int k = 0; k < 64; ++k)
              acc = fmaf(sWUOT[k * 128 + tid], sMix[k], acc);
            #pragma unroll 8
            for (int k = 0; k < 64; ++k)
              acc = fmaf(sWUOT[(64 + k) * 128 + tid], sCT[k * STRIDE + i], acc);
            sG[tid] = acc;
          }
          __syncthreads();

          // step 5: new state column
          if (tid < 64) {
            const float u = 1.f / (1.f + __expf(-sG[tid]));
            const float o = fmaxf(sG[64 + tid], 0.f);
            const float ns = u * o + (1.f - u) * sMix[tid];
            sNew[tid] = ns;
            sCT[tid * STRIDE + i] = ns;
          }
          __syncthreads();

          // step 6: incremental column update of YCA/YCB = W_C*_L @ C (branchless store)
          if (tid < 128) {
            float acc = 0.f;
            #pragma unroll 8
            for (int k = 0; k < 64; ++k)
              acc = fmaf(sWCLT[k * 128 + tid], sNew[k], acc);
            const int ybase = (2 + (tid >> 6)) * FS;     // 2=YCA, 3=YCB
            sY[ybase + (tid & 63) * STRIDE + i] = acc;
          }
          __syncthreads();
        }

        // ---- write level back ----
        for (int idx = tid; idx < FD * PD; idx += NT) {
          int f = idx >> 6, p = idx & 63;
          ws[lc * 4096 + idx] = sCT[f * STRIDE + p];
        }
        __syncthreads();
      }
    }
  }

  // ---------------- final projection: out.T = W2 @ (W1 @ states + b1) + b2 ----------------
  for (int n = tid; n < LV * PD; n += NT) {
    const int l = n >> 6, p = n & 63;
    float h[10];
    #pragma unroll
    for (int j = 0; j < 10; ++j) {
      float a = b1[j];
      #pragma unroll 8
      for (int f = 0; f < 64; ++f)
        a = fmaf(W1[j * 64 + f], ws[l * 4096 + f * 64 + p], a);
      h[j] = a;
    }
    #pragma unroll
    for (int o = 0; o < 16; ++o) {
      float a = b2[o];
      #pragma unroll
      for (int j = 0; j < 10; ++j) a = fmaf(W2[o * 10 + j], h[j], a);
      out[o * 1024 + n] = a;
    }
  }
}

extern "C" void kernel_launch(void* const* d_in, const int* in_sizes, int n_in,
                              void* d_out, int out_size, void* d_ws, size_t ws_size,
                              hipStream_t stream) {
  (void)in_sizes; (void)n_in; (void)out_size; (void)ws_size;
  const float* x    = (const float*)d_in[0];
  const float* W_TA = (const float*)d_in[1];
  const float* b_TA = (const float*)d_in[2];
  const float* W_CA = (const float*)d_in[3];
  const float* b_CA = (const float*)d_in[4];
  const float* W_TB = (const float*)d_in[5];
  const float* b_TB = (const float*)d_in[6];
  const float* W_CB = (const float*)d_in[7];
  const float* b_CB = (const float*)d_in[8];
  const float* Wmix = (const float*)d_in[9];
  const float* W_u  = (const float*)d_in[10];
  const float* b_u  = (const float*)d_in[11];
  const float* W_o  = (const float*)d_in[12];
  const float* b_o  = (const float*)d_in[13];
  const float* W1   = (const float*)d_in[14];
  const float* b1   = (const float*)d_in[15];
  const float* W2   = (const float*)d_in[16];
  const float* b2   = (const float*)d_in[17];
  const int*   np   = (const int*)d_in[18];

  float* ws  = (float*)d_ws;     // needs 16*64*64*4 = 256 KB
  float* out = (float*)d_out;

  WAVE_34900904247805_kernel<<<dim3(1), dim3(NT), 0, stream>>>(
      x, W_TA, b_TA, W_CA, b_CA, W_TB, b_TB, W_CB, b_CB, Wmix,
      W_u, b_u, W_o, b_o, W1, b1, W2, b2, np, ws, out);
}